// DiagonalMicroAttention_13855564497421
// MI455X (gfx1250) — compile-verified
//
#include <hip/hip_runtime.h>
#include <hip/hip_bf16.h>
#include <math.h>

typedef float v2f __attribute__((ext_vector_type(2)));
typedef float v4f __attribute__((ext_vector_type(4)));
typedef float v8f __attribute__((ext_vector_type(8)));
typedef int   v4i __attribute__((ext_vector_type(4)));

// Native FP32 matrix op on CDNA5: D(16x16) = A(16x4) * B(4x16) + C
__device__ __forceinline__ v8f wmma_f32(v2f a, v2f b, v8f c) {
  return __builtin_amdgcn_wmma_f32_16x16x4_f32(false, a, false, b, (short)0, c,
                                               false, false);
}

// ---- gfx1250 async global->LDS copy (ASYNCcnt path), with safe fallback ----
#if __has_builtin(__builtin_amdgcn_global_load_async_to_lds_b128)
#define HAVE_ASYNC_LDS 1
#endif

__device__ __forceinline__ void async_copy16(const float* g, float* l) {
#ifdef HAVE_ASYNC_LDS
  __builtin_amdgcn_global_load_async_to_lds_b128(
      (__attribute__((address_space(1))) v4i*)g,
      (__attribute__((address_space(3))) v4i*)l, 0, 0);
#else
  *(v4f*)l = *(const v4f*)g;   // VGPR round-trip fallback
#endif
}

__device__ __forceinline__ void async_wait0() {
#ifdef HAVE_ASYNC_LDS
#if __has_builtin(__builtin_amdgcn_s_wait_asynccnt)
  __builtin_amdgcn_s_wait_asynccnt(0);
#else
  asm volatile("s_wait_asynccnt 0x0" ::: "memory");
#endif
#endif
}

// ---------------------------------------------------------------------------
// Kernel 1: qkv[b,o,p] = sum_c w_qkv[o,c] * x[b,c,p]   (GEMM 768x256 x 256x1024)
// scattered into q/k/v stored [b, h, n, d] with d contiguous.
// ---------------------------------------------------------------------------
__global__ __launch_bounds__(32)
void qkv_gemm_kernel(const float* __restrict__ x, const float* __restrict__ w_qkv,
                     float* __restrict__ q, float* __restrict__ k,
                     float* __restrict__ v) {
  const int ptile = blockIdx.x * 16;
  const int otile = blockIdx.y * 16;
  const int b     = blockIdx.z;
  const int lane  = threadIdx.x;
  const int col   = lane & 15;
  const int hi    = lane >> 4;
  const int koff  = hi * 2;
  const float* xb = x + (size_t)b * 256 * 1024;

  v8f acc = {};
  for (int k0 = 0; k0 < 256; k0 += 4) {
    __builtin_prefetch(xb + (k0 + 4 + koff) * 1024 + ptile + col, 0, 3);
    v2f a = *(const v2f*)(w_qkv + (otile + col) * 256 + k0 + koff);
    v2f bf;
    bf.x = xb[(k0 + koff)     * 1024 + ptile + col];
    bf.y = xb[(k0 + koff + 1) * 1024 + ptile + col];
    acc = wmma_f32(a, bf, acc);
  }

  const int which = otile >> 8;
  float* base = (which == 0) ? q : ((which == 1) ? k : v);
  const int obase = otile & 255;
#pragma unroll
  for (int r = 0; r < 8; ++r) {
    int c  = obase + r + hi * 8;
    int h  = c >> 5;
    int d  = c & 31;
    int p  = ptile + col;
    base[(((size_t)b * 8 + h) * 1024 + p) * 32 + d] = acc[r];
  }
}

// ---------------------------------------------------------------------------
// Kernel 2a: vsum[b,h,d] = sum_j v[b,h,j,d]
// ---------------------------------------------------------------------------
__global__ __launch_bounds__(32)
void vsum_kernel(const float* __restrict__ v, float* __restrict__ vsum) {
  const int bh = blockIdx.x;
  const int d  = threadIdx.x;
  const float* vp = v + (size_t)bh * 1024 * 32 + d;
  float s = 0.f;
  for (int j = 0; j < 1024; ++j) s += vp[j * 32];
  vsum[bh * 32 + d] = s;
}

// ---------------------------------------------------------------------------
// Kernel 2b: diag_vec[b,t] = (scale/8) * sum_h sum_pos q[b,h,33*pos,t]*k[b,h,33*pos,t]
// ---------------------------------------------------------------------------
__global__ __launch_bounds__(32)
void diag_kernel(const float* __restrict__ q, const float* __restrict__ k,
                 float* __restrict__ diag) {
  const int b = blockIdx.x;
  const int t = threadIdx.x;
  float acc = 0.f;
  for (int h = 0; h < 8; ++h) {
    const float* qp = q + (((size_t)b * 8 + h) * 1024) * 32;
    const float* kp = k + (((size_t)b * 8 + h) * 1024) * 32;
    for (int pos = 0; pos < 32; ++pos) {
      int n = pos * 33;
      acc += qp[n * 32 + t] * kp[n * 32 + t];
    }
  }
  const float scale = 0.17677669529663687f;
  diag[b * 32 + t] = acc * scale * 0.125f;
}

// ---------------------------------------------------------------------------
// Kernel 3: flash attention per (b,h). 4 waves / 64 query rows per workgroup;
// K/V j-tiles staged once per workgroup via GLOBAL_LOAD_ASYNC_TO_LDS and
// consumed from LDS by all 4 waves (4x cut of global K/V traffic).
// ---------------------------------------------------------------------------
__global__ __launch_bounds__(128)
void flash_attn_kernel(const float* __restrict__ q, const float* __restrict__ k,
                       const float* __restrict__ v, const float* __restrict__ vsum,
                       const float* __restrict__ diag,
                       float* __restrict__ attnout) {
  __shared__ float ktile[16][32];        // K rows j0..j0+15, d contiguous
  __shared__ float vtile[16][32];        // V rows j0..j0+15
  __shared__ float pst[4][16][17];       // per-wave P staging (D- -> A-layout)

  const int tid  = threadIdx.x;
  const int wid  = tid >> 5;
  const int lane = tid & 31;
  const int col  = lane & 15;
  const int hi   = lane >> 4;
  const int koff = hi * 2;
  const int i0   = blockIdx.x * 64 + wid * 16;
  const int h    = blockIdx.y;
  const int b    = blockIdx.z;
  const int bh   = b * 8 + h;
  const float scale = 0.17677669529663687f;

  const float* Q = q + (size_t)bh * 1024 * 32;
  const float* K = k + (size_t)bh * 1024 * 32;
  const float* V = v + (size_t)bh * 1024 * 32;

  // cooperative-copy coordinates: 128 threads x 16B = one 16x32 f32 tile
  const int crow = tid >> 3;             // 0..15
  const int cseg = (tid & 7) * 4;        // 0,4,..,28

  // A-fragments of Q (16 rows x d=32) held in registers for the whole pass
  v2f qa[8];
#pragma unroll
  for (int kk = 0; kk < 8; ++kk)
    qa[kk] = *(const v2f*)(Q + (i0 + col) * 32 + kk * 4 + koff);

  v8f o0 = {}, o1 = {};
  float m[8], l[8];
#pragma unroll
  for (int r = 0; r < 8; ++r) { m[r] = -1e30f; l[r] = 0.f; }

  for (int jt = 0; jt < 64; ++jt) {
    const int j0 = jt * 16;
    // ---- stage K/V tile into LDS (ASYNCcnt-tracked, no VGPR round trip) ----
    async_copy16(K + (j0 + crow) * 32 + cseg, &ktile[crow][cseg]);
    async_copy16(V + (j0 + crow) * 32 + cseg, &vtile[crow][cseg]);
    async_wait0();
    __syncthreads();

    // ---- S tile (16x16) = Q(16x32) * K^T(32x16) ----
    v8f s = {};
#pragma unroll
    for (int kk = 0; kk < 8; ++kk) {
      v2f kb = *(const v2f*)&ktile[col][kk * 4 + koff];
      s = wmma_f32(qa[kk], kb, s);
    }
    // ---- online softmax (width-16 xor-shuffle row reductions) ----
#pragma unroll
    for (int r = 0; r < 8; ++r) {
      float sv = s[r] * scale;
      float tm = sv;
#pragma unroll
      for (int mk = 1; mk < 16; mk <<= 1)
        tm = fmaxf(tm, __shfl_xor(tm, mk, 16));
      float mn    = fmaxf(m[r], tm);
      float alpha = __expf(m[r] - mn);
      float p     = __expf(sv - mn);
      float ts = p;
#pragma unroll
      for (int mk = 1; mk < 16; mk <<= 1)
        ts += __shfl_xor(ts, mk, 16);
      l[r] = l[r] * alpha + ts;
      m[r] = mn;
      o0[r] *= alpha;
      o1[r] *= alpha;
      pst[wid][r + hi * 8][col] = p;     // same-wave LDS: DS ops stay in order
    }
    // ---- O(16x32) += P(16x16) * V(16x32) ----
#pragma unroll
    for (int kk = 0; kk < 4; ++kk) {
      v2f pa;
      pa.x = pst[wid][col][kk * 4 + koff];
      pa.y = pst[wid][col][kk * 4 + koff + 1];
      v2f vb0, vb1;
      vb0.x = vtile[kk * 4 + koff][col];
      vb0.y = vtile[kk * 4 + koff + 1][col];
      vb1.x = vtile[kk * 4 + koff][16 + col];
      vb1.y = vtile[kk * 4 + koff + 1][16 + col];
      o0 = wmma_f32(pa, vb0, o0);
      o1 = wmma_f32(pa, vb1, o1);
    }
    __syncthreads();                     // protect k/v tiles before next stage
  }

#pragma unroll
  for (int r = 0; r < 8; ++r) {
    int   i    = i0 + r + hi * 8;
    float inv  = 1.0f / l[r];
    float coef = (i < 32) ? 0.3f * diag[b * 32 + i] : 0.0f;
    float out0 = o0[r] * inv + coef * vsum[bh * 32 + col];
    float out1 = o1[r] * inv + coef * vsum[bh * 32 + 16 + col];
    attnout[((size_t)b * 256 + h * 32 + col)      * 1024 + i] = out0;
    attnout[((size_t)b * 256 + h * 32 + 16 + col) * 1024 + i] = out1;
  }
}

// ---------------------------------------------------------------------------
// Kernel 4: gate[b,p] = sigmoid(w_g2 . gelu(W_g1 . pooled[b,:,p] + b_g1) + b_g2)
// ---------------------------------------------------------------------------
__global__ __launch_bounds__(256)
void gate_kernel(const float* __restrict__ x, const float* __restrict__ w_g1,
                 const float* __restrict__ b_g1, const float* __restrict__ w_g2,
                 const float* __restrict__ b_g2, float* __restrict__ gate) {
  __shared__ float pooled[256][32];
  __shared__ float gact[64][32];
  const int pg  = blockIdx.x;
  const int b   = blockIdx.y;
  const int tid = threadIdx.x;
  const float* xb = x + (size_t)b * 256 * 1024;

  for (int idx = tid; idx < 256 * 32; idx += 256) {
    int c = idx >> 5, pix = idx & 31;
    int p = pg * 32 + pix;
    int hrow = p >> 5, wcol = p & 31;
    int aw = wcol >> 1;
    float va = xb[c * 1024 + hrow * 32 + aw];
    float vb = xb[c * 1024 + hrow * 32 + (31 - aw)];
    pooled[c][pix] = fabsf(va - vb);
  }
  __syncthreads();
  for (int t = tid; t < 64 * 32; t += 256) {
    int oc = t >> 5, pix = t & 31;
    float s = b_g1[oc];
    for (int c = 0; c < 256; ++c) s += w_g1[oc * 256 + c] * pooled[c][pix];
    s = 0.5f * s * (1.0f + erff(s * 0.70710678118654752f));
    gact[oc][pix] = s;
  }
  __syncthreads();
  if (tid < 32) {
    float s = b_g2[0];
    for (int oc = 0; oc < 64; ++oc) s += w_g2[oc] * gact[oc][tid];
    gate[b * 1024 + pg * 32 + tid] = 1.0f / (1.0f + __expf(-s));
  }
}

// ---------------------------------------------------------------------------
// Kernel 5: out[b,o,p] = sum_c w_out[o,c] * (attnout[b,c,p]*(1+g[b,p])) + b_out[o]
// ---------------------------------------------------------------------------
__global__ __launch_bounds__(32)
void out_gemm_kernel(const float* __restrict__ attnout,
                     const float* __restrict__ gate,
                     const float* __restrict__ w_out,
                     const float* __restrict__ b_out,
                     float* __restrict__ out) {
  const int ptile = blockIdx.x * 16;
  const int otile = blockIdx.y * 16;
  const int b     = blockIdx.z;
  const int lane  = threadIdx.x;
  const int col   = lane & 15;
  const int hi    = lane >> 4;
  const int koff  = hi * 2;
  const float* yb = attnout + (size_t)b * 256 * 1024;
  const float gmul = 1.0f + gate[b * 1024 + ptile + col];

  v8f acc = {};
  for (int k0 = 0; k0 < 256; k0 += 4) {
    __builtin_prefetch(yb + (k0 + 4 + koff) * 1024 + ptile + col, 0, 3);
    v2f a = *(const v2f*)(w_out + (otile + col) * 256 + k0 + koff);
    v2f bf;
    bf.x = yb[(k0 + koff)     * 1024 + ptile + col] * gmul;
    bf.y = yb[(k0 + koff + 1) * 1024 + ptile + col] * gmul;
    acc = wmma_f32(a, bf, acc);
  }
#pragma unroll
  for (int r = 0; r < 8; ++r) {
    int o = otile + r + hi * 8;
    out[((size_t)b * 256 + o) * 1024 + ptile + col] = acc[r] + b_out[o];
  }
}

// ---------------------------------------------------------------------------
extern "C" void kernel_launch(void* const* d_in, const int* in_sizes, int n_in,
                              void* d_out, int out_size, void* d_ws, size_t ws_size,
                              hipStream_t stream) {
  const float* x     = (const float*)d_in[0];
  const float* w_qkv = (const float*)d_in[1];
  const float* w_out = (const float*)d_in[2];
  const float* b_out = (const float*)d_in[3];
  const float* w_g1  = (const float*)d_in[4];
  const float* b_g1  = (const float*)d_in[5];
  const float* w_g2  = (const float*)d_in[6];
  const float* b_g2  = (const float*)d_in[7];
  float* out = (float*)d_out;

  float* ws      = (float*)d_ws;
  float* q       = ws;                    // 8*8*1024*32 floats
  float* k       = q + 2097152;
  float* v       = k + 2097152;
  float* attnout = v + 2097152;           // [b][256][1024]
  float* vsum    = attnout + 2097152;     // [b*8][32]
  float* diag    = vsum + 2048;           // [b][32]
  float* gate    = diag + 256;            // [b][1024]

  qkv_gemm_kernel<<<dim3(64, 48, 8), 32, 0, stream>>>(x, w_qkv, q, k, v);
  vsum_kernel    <<<dim3(64),        32, 0, stream>>>(v, vsum);
  diag_kernel    <<<dim3(8),         32, 0, stream>>>(q, k, diag);
  gate_kernel    <<<dim3(32, 8),    256, 0, stream>>>(x, w_g1, b_g1, w_g2, b_g2, gate);
  flash_attn_kernel<<<dim3(16, 8, 8), 128, 0, stream>>>(q, k, v, vsum, diag, attnout);
  out_gemm_kernel<<<dim3(64, 16, 8), 32, 0, stream>>>(attnout, gate, w_out, b_out, out);
}